// LSTMModel_29231547416885
// MI455X (gfx1250) — compile-verified
//
#include <hip/hip_runtime.h>
#include <hip/hip_bf16.h>

typedef __attribute__((ext_vector_type(2))) float v2f;
typedef __attribute__((ext_vector_type(8))) float v8f;

#define SEQ_T 4096
#define HID   150

// ---------------------------------------------------------------------------
// Generic strided GEMM:  C[M x N] = A (.) B^T + bias1 + bias2
//   C[m][n] = sum_k A[m*saM + k*saK] * B[n*sbN + k*sbK]
// Each wave computes a 16x64 C strip: one A fetch feeds 4
// V_WMMA_F32_16X16X4_F32 ops (exact fp32 matrix path).
// AK1/BK1 template flags: K-contiguous operands use true 8B vector loads.
// Requirements: M % 128 == 0, K % 4 == 0. N edge-masked without divergence
// (clamped address + multiply-by-0) so EXEC is all-1s at every WMMA.
// ---------------------------------------------------------------------------
template <bool AK1, bool BK1>
__global__ void gemm_wmma_nt(const float* __restrict__ A,
                             const float* __restrict__ B,
                             float* __restrict__ C,
                             const float* __restrict__ bias1,
                             const float* __restrict__ bias2,
                             int M, int N, int K,
                             long saM, long saK, long sbN, long sbK)
{
    const int lane = threadIdx.x & 31;
    const int wave = threadIdx.x >> 5;
    const int tm   = (blockIdx.y * 8 + wave) * 16;   // tile row base (always < M)
    const int tn   = blockIdx.x * 64;                // strip col base (may exceed N)
    const int half = lane >> 4;                      // 0: K pair {0,1}, 1: {2,3}
    const int l16  = lane & 15;

    const int am = tm + l16;                         // A row for this lane (valid)
    const float* __restrict__ Ap = A + (long)am * saM;

    const float* __restrict__ Bp[4];
    float bscale[4];
#pragma unroll
    for (int i = 0; i < 4; ++i) {
        const int bn = tn + i * 16 + l16;            // B row (== C col) this lane
        bscale[i] = (bn < N) ? 1.0f : 0.0f;
        Bp[i] = B + (long)((bn < N) ? bn : 0) * sbN;
    }

    v8f acc[4];
#pragma unroll
    for (int i = 0; i < 4; ++i) acc[i] = v8f{0.f,0.f,0.f,0.f,0.f,0.f,0.f,0.f};

    for (int k0 = 0; k0 < K; k0 += 4) {
        const long ka = (long)k0 + 2 * half;
        v2f a;
        if (AK1) {
            a = *(const v2f*)&Ap[ka];                // global_load_b64
        } else {
            a.x = Ap[ka * saK];
            a.y = Ap[(ka + 1) * saK];
        }
#pragma unroll
        for (int i = 0; i < 4; ++i) {
            v2f b;
            if (BK1) {
                b = *(const v2f*)&Bp[i][ka];         // global_load_b64
            } else {
                b.x = Bp[i][ka * sbK];
                b.y = Bp[i][(ka + 1) * sbK];
            }
            b.x *= bscale[i];
            b.y *= bscale[i];
            acc[i] = __builtin_amdgcn_wmma_f32_16x16x4_f32(
                         false, a, false, b, (short)0, acc[i], false, false);
        }
    }

#pragma unroll
    for (int i = 0; i < 4; ++i) {
        const int col = tn + i * 16 + l16;
        if (col < N) {
            float bias = 0.0f;
            if (bias1) bias += bias1[col];
            if (bias2) bias += bias2[col];
#pragma unroll
            for (int v = 0; v < 8; ++v) {
                const int row = tm + v + 8 * half;   // C layout per ISA 7.12.2
                C[(size_t)row * N + col] = acc[i][v] + bias;
            }
        }
    }
}

// ---------------------------------------------------------------------------
// Sequential LSTM recurrence for one layer. 4 blocks: (model, dir).
// 640 threads: threads 0..599 each own one gate row of Whh (150 weights,
// L2/WGP$-resident); threads 0..149 own the (c,h) update for one hidden unit.
// h lives in LDS; c lives in a register of its owning thread.
// xg layout per (m,d): [T][600] (gate order i,f,g,o). out: [T][300], fwd in
// cols 0..149, bwd in cols 150..299 at the un-reversed time index.
// ---------------------------------------------------------------------------
__global__ void lstm_recurrence(const float* __restrict__ xg_base,
                                float* __restrict__ out_base,
                                const float* __restrict__ whh_solv,
                                const float* __restrict__ whh_solu,
                                int layer, int T)
{
    const int m = blockIdx.x >> 1;
    const int d = blockIdx.x & 1;

    const float* __restrict__ Whh =
        (m ? whh_solu : whh_solv) + (size_t)(layer * 2 + d) * 600 * 150;
    const float* __restrict__ xg  = xg_base + (size_t)(m * 2 + d) * T * 600;
    float* __restrict__ out       = out_base + (size_t)m * T * 300;

    __shared__ __align__(16) float h[160];
    __shared__ float gact[600];

    const int j = threadIdx.x;
    float c = 0.0f;                         // hidden-cell state, threads j<150

    if (j < 160) h[j] = 0.0f;
    __syncthreads();

    const float* __restrict__ wrow = Whh + (size_t)j * 150; // j<600 only used

    for (int s = 0; s < T; ++s) {
        const int t = d ? (T - 1 - s) : s;

        if (j < 600) {
            float dot = 0.0f;
#pragma unroll 5
            for (int k = 0; k < 150; k += 2) {          // 8B-aligned pairs
                const float2 hv = *(const float2*)&h[k];
                const float2 wv = *(const float2*)&wrow[k];
                dot = fmaf(hv.x, wv.x, dot);
                dot = fmaf(hv.y, wv.y, dot);
            }
            const float g = xg[(size_t)t * 600 + j] + dot;
            float a;
            if (j >= 300 && j < 450) a = tanhf(g);              // g-gate
            else                     a = 1.0f / (1.0f + __expf(-g)); // i,f,o
            gact[j] = a;
        }
        __syncthreads();

        if (j < 150) {
            c = fmaf(gact[150 + j], c, gact[j] * gact[300 + j]); // f*c + i*g
            const float hn = gact[450 + j] * tanhf(c);           // o*tanh(c)
            h[j] = hn;
            out[(size_t)t * 300 + d * 150 + j] = hn;
        }
        __syncthreads();
    }
}

// ---------------------------------------------------------------------------
// Row softmax, in place. One block per row; S stays L2-resident (64 MB).
// ---------------------------------------------------------------------------
__global__ void softmax_rows(float* __restrict__ S, int T)
{
    float* row = S + (size_t)blockIdx.x * T;
    __shared__ float red[256];
    const int t = threadIdx.x;

    float mx = -3.4e38f;
    for (int i = t; i < T; i += 256) mx = fmaxf(mx, row[i]);
    red[t] = mx; __syncthreads();
    for (int sh = 128; sh > 0; sh >>= 1) {
        if (t < sh) red[t] = fmaxf(red[t], red[t + sh]);
        __syncthreads();
    }
    mx = red[0]; __syncthreads();

    float s = 0.0f;
    for (int i = t; i < T; i += 256) {
        const float e = __expf(row[i] - mx);
        row[i] = e;
        s += e;
    }
    red[t] = s; __syncthreads();
    for (int sh = 128; sh > 0; sh >>= 1) {
        if (t < sh) red[t] += red[t + sh];
        __syncthreads();
    }
    const float inv = 1.0f / red[0];
    for (int i = t; i < T; i += 256) row[i] *= inv;
}

// ---------------------------------------------------------------------------
// inp[b] = sum_i max(H[i][b], P[i][b])          for b in [0,300)
// inp[b] = sum_i max(G[i][b-300], Q[i][b-300])  for b in [300,600)
// ---------------------------------------------------------------------------
__global__ void col_max_sum(const float* __restrict__ H, const float* __restrict__ P,
                            const float* __restrict__ G, const float* __restrict__ Q,
                            float* __restrict__ inp, int T)
{
    __shared__ float red[256];
    const int b = blockIdx.x;
    const int t = threadIdx.x;
    const float* X = (b < 300) ? H : G;
    const float* Y = (b < 300) ? P : Q;
    const int c = (b < 300) ? b : b - 300;

    float s = 0.0f;
    for (int i = t; i < T; i += 256)
        s += fmaxf(X[(size_t)i * 300 + c], Y[(size_t)i * 300 + c]);
    red[t] = s; __syncthreads();
    for (int sh = 128; sh > 0; sh >>= 1) {
        if (t < sh) red[t] += red[t + sh];
        __syncthreads();
    }
    if (t == 0) inp[b] = red[0];
}

// out[j] = relu(dot(inp[600], W[j]) + b[j]),  one block per j
__global__ void fc1_kernel(const float* __restrict__ inp, const float* __restrict__ W,
                           const float* __restrict__ bias, float* __restrict__ out)
{
    __shared__ float red[128];
    const int j = blockIdx.x;
    const int t = threadIdx.x;
    const float* w = W + (size_t)j * 600;
    float s = 0.0f;
    for (int k = t; k < 600; k += 128) s = fmaf(inp[k], w[k], s);
    red[t] = s; __syncthreads();
    for (int sh = 64; sh > 0; sh >>= 1) {
        if (t < sh) red[t] += red[t + sh];
        __syncthreads();
    }
    if (t == 0) out[j] = fmaxf(red[0] + bias[j], 0.0f);
}

// out[0] = dot(x[2000], w[2000]) + b[0]
__global__ void fc2_kernel(const float* __restrict__ x, const float* __restrict__ w,
                           const float* __restrict__ bias, float* __restrict__ out)
{
    __shared__ float red[256];
    const int t = threadIdx.x;
    float s = 0.0f;
    for (int k = t; k < 2000; k += 256) s = fmaf(x[k], w[k], s);
    red[t] = s; __syncthreads();
    for (int sh = 128; sh > 0; sh >>= 1) {
        if (t < sh) red[t] += red[t + sh];
        __syncthreads();
    }
    if (t == 0) out[0] = red[0] + bias[0];
}

// ---------------------------------------------------------------------------
extern "C" void kernel_launch(void* const* d_in, const int* in_sizes, int n_in,
                              void* d_out, int out_size, void* d_ws, size_t ws_size,
                              hipStream_t stream)
{
    (void)in_sizes; (void)n_in; (void)out_size; (void)ws_size;

    const float* x_solv   = (const float*)d_in[0];   // [1,4096,300]
    const float* x_solu   = (const float*)d_in[1];
    const float* solv_Wih = (const float*)d_in[2];   // [2,2,600,300]
    const float* solv_Whh = (const float*)d_in[3];   // [2,2,600,150]
    const float* solv_bih = (const float*)d_in[4];   // [2,2,600]
    const float* solv_bhh = (const float*)d_in[5];
    const float* solu_Wih = (const float*)d_in[6];
    const float* solu_Whh = (const float*)d_in[7];
    const float* solu_bih = (const float*)d_in[8];
    const float* solu_bhh = (const float*)d_in[9];
    const float* fc1_w    = (const float*)d_in[10];  // [2000,600]
    const float* fc1_b    = (const float*)d_in[11];
    const float* fc2_w    = (const float*)d_in[12];  // [1,2000]
    const float* fc2_b    = (const float*)d_in[13];
    float* out            = (float*)d_out;

    const int T = SEQ_T;
    float* ws = (float*)d_ws;

    // workspace layout (floats)
    const size_t XG_OFF   = 0;                                   // 4 * T * 600
    const size_t OUT0_OFF = XG_OFF   + 4ull * T * 600;           // 2 * T * 300
    const size_t OUT1_OFF = OUT0_OFF + 2ull * T * 300;           // 2 * T * 300
    const size_t S_OFF    = OUT1_OFF + 2ull * T * 300;           // T * T
    const size_t P_OFF    = S_OFF    + (size_t)T * T;            // T * 300
    const size_t Q_OFF    = P_OFF    + (size_t)T * 300;          // T * 300
    const size_t INP_OFF  = Q_OFF    + (size_t)T * 300;          // 600
    const size_t FC1_OFF  = INP_OFF  + 600;                      // 2000

    float* XG   = ws + XG_OFF;
    float* OUT0 = ws + OUT0_OFF;
    float* OUT1 = ws + OUT1_OFF;
    float* S    = ws + S_OFF;
    float* P    = ws + P_OFF;
    float* Q    = ws + Q_OFF;
    float* INP  = ws + INP_OFF;
    float* FC1  = ws + FC1_OFF;

    const dim3 gblock(256);
    auto gemm = [&](const float* A, const float* B, float* C,
                    const float* b1, const float* b2,
                    int M, int N, int K,
                    long saM, long saK, long sbN, long sbK) {
        dim3 grid((N + 63) / 64, M / 128);
        if (saK == 1 && sbK == 1) {
            hipLaunchKernelGGL((gemm_wmma_nt<true, true>), grid, gblock, 0, stream,
                               A, B, C, b1, b2, M, N, K, saM, saK, sbN, sbK);
        } else if (saK == 1) {
            hipLaunchKernelGGL((gemm_wmma_nt<true, false>), grid, gblock, 0, stream,
                               A, B, C, b1, b2, M, N, K, saM, saK, sbN, sbK);
        } else {
            hipLaunchKernelGGL((gemm_wmma_nt<false, false>), grid, gblock, 0, stream,
                               A, B, C, b1, b2, M, N, K, saM, saK, sbN, sbK);
        }
    };

    // ---- layer 0: xg = x @ Wih[0,d].T + (bih+bhh) for both models/dirs ----
    for (int m = 0; m < 2; ++m) {
        const float* X   = m ? x_solu : x_solv;
        const float* Wih = m ? solu_Wih : solv_Wih;
        const float* bi  = m ? solu_bih : solv_bih;
        const float* bh  = m ? solu_bhh : solv_bhh;
        for (int d = 0; d < 2; ++d) {
            gemm(X, Wih + (size_t)d * 600 * 300,
                 XG + (size_t)(m * 2 + d) * T * 600,
                 bi + (size_t)d * 600, bh + (size_t)d * 600,
                 T, 600, 300, 300, 1, 300, 1);
        }
    }
    hipLaunchKernelGGL(lstm_recurrence, dim3(4), dim3(640), 0, stream,
                       XG, OUT0, solv_Whh, solu_Whh, 0, T);

    // ---- layer 1: input is concat(fwd,bwd) = OUT0[m] (T x 300) ----
    for (int m = 0; m < 2; ++m) {
        const float* Wih = (m ? solu_Wih : solv_Wih) + 2ull * 600 * 300;
        const float* bi  = (m ? solu_bih : solv_bih) + 2ull * 600;
        const float* bh  = (m ? solu_bhh : solv_bhh) + 2ull * 600;
        for (int d = 0; d < 2; ++d) {
            gemm(OUT0 + (size_t)m * T * 300, Wih + (size_t)d * 600 * 300,
                 XG + (size_t)(m * 2 + d) * T * 600,
                 bi + (size_t)d * 600, bh + (size_t)d * 600,
                 T, 600, 300, 300, 1, 300, 1);
        }
    }
    hipLaunchKernelGGL(lstm_recurrence, dim3(4), dim3(640), 0, stream,
                       XG, OUT1, solv_Whh, solu_Whh, 1, T);

    const float* H = OUT1;                    // [T,300]
    const float* G = OUT1 + (size_t)T * 300;  // [T,300]

    // ---- attention ----
    // S = H @ G^T
    gemm(H, G, S, nullptr, nullptr, T, T, 300, 300, 1, 300, 1);
    // a = softmax(S, axis=1), in place
    hipLaunchKernelGGL(softmax_rows, dim3(T), dim3(256), 0, stream, S, T);
    // P = a @ G   : B[n][k] = G[k][n]  -> sbN=1, sbK=300
    gemm(S, G, P, nullptr, nullptr, T, 300, T, (long)T, 1, 1, 300);
    // Q = a^T @ H : A[m][k] = a[k][m]  -> saM=1, saK=T ; B[n][k] = H[k][n]
    gemm(S, H, Q, nullptr, nullptr, T, 300, T, 1, (long)T, 1, 300);

    // inp = concat(sum_i max(H,P), sum_i max(G,Q))
    hipLaunchKernelGGL(col_max_sum, dim3(600), dim3(256), 0, stream,
                       H, P, G, Q, INP, T);

    // ---- MLP head ----
    hipLaunchKernelGGL(fc1_kernel, dim3(2000), dim3(128), 0, stream,
                       INP, fc1_w, fc1_b, FC1);
    hipLaunchKernelGGL(fc2_kernel, dim3(1), dim3(256), 0, stream,
                       FC1, fc2_w, fc2_b, out);
}